// ContrastiveCorrelationLoss_43808666419315
// MI455X (gfx1250) — compile-verified
//
#include <hip/hip_runtime.h>
#include <math.h>

// ContrastiveCorrelationLoss for MI455X (gfx1250, wave32).
// Kernel 1: one wave per sample point (2*32*11*11 = 7744 waves).
//   - 16 channels/lane; adjacent-pixel taps gathered as 64-bit pairs.
//   - wave-wide reductions via V_WMMA_F32_16X16X4_F32 (B = ones trick)
//     + one ds_swizzle SWAPX16 to combine the two wave halves.
// Kernel 2: deterministic single-block tree reduction to the scalar output.

typedef __attribute__((ext_vector_type(2))) float v2f;
typedef __attribute__((ext_vector_type(8))) float v8f;

// 4-byte-aligned float pair: backend emits global_load_b64 when the target's
// unaligned-access mode allows it, otherwise two b32 loads. Correct either way.
struct __attribute__((packed, aligned(4))) f2u { float x, y; };

#define CCL_B 32
#define CCL_C 512
#define CCL_H 56
#define CCL_W 56
#define CCL_S 11
#define CCL_HW (CCL_H * CCL_W)
#define CCL_NPTS (CCL_B * CCL_S * CCL_S)   // 3872 per branch
#define CCL_TOTAL (2 * CCL_NPTS)           // 7744 waves
#define POS_W 0.577453483136995f
#define NEG_W 0.9058762625226623f

// Full-wave (32-lane) sum using v_wmma_f32_16x16x4_f32.
// A layout (32-bit 16x4): VGPR0 -> lanes 0-15: (M=lane, K=0); lanes 16-31: (M=lane-16, K=2)
//                         VGPR1 -> K=1 / K=3  (we set it to 0)
// B = all ones (identical in every slot -> layout-independent), C = 0.
// => D[m][n] = v_m + v_{m+16} for all n.
// D layout: VGPR j -> lanes 0-15: M=j ; lanes 16-31: M=j+8 (value independent of N).
// Local sum of the 8 D regs gives rows 0-7 (low half) / rows 8-15 (high half);
// ds_swizzle SWAPX16 (xor_mask=0x10, and_mask=0x1f -> imm 0x401F) swaps halves
// so one add completes the 32-lane reduction on every lane.
__device__ __forceinline__ float wave_reduce_wmma(float v) {
  v2f a;
  a.x = v;
  a.y = 0.0f;
  v2f b;
  b.x = 1.0f;
  b.y = 1.0f;
  v8f c = {};
  v8f d = __builtin_amdgcn_wmma_f32_16x16x4_f32(
      /*neg_a=*/false, a, /*neg_b=*/false, b,
      /*c_mod=*/(short)0, c, /*reuse_a=*/false, /*reuse_b=*/false);
  float local = ((d[0] + d[1]) + (d[2] + d[3])) + ((d[4] + d[5]) + (d[6] + d[7]));
  int other = __builtin_amdgcn_ds_swizzle(__float_as_int(local), 0x401F);
  return local + __int_as_float(other);
}

__global__ __launch_bounds__(256) void ccl_point_kernel(
    const float* __restrict__ orig_feats, const float* __restrict__ orig_feats_pos,
    const float* __restrict__ orig_code,
    const float* __restrict__ nega_feats, const float* __restrict__ nega_feats_pos,
    const float* __restrict__ nega_code,
    const float* __restrict__ coords1, const float* __restrict__ coords2,
    float* __restrict__ ws) {
  const int wid  = (int)((blockIdx.x * blockDim.x + threadIdx.x) >> 5);
  const int lane = (int)(threadIdx.x & 31);

  const bool neg = (wid >= CCL_NPTS);
  const int p = neg ? (wid - CCL_NPTS) : wid;

  const float* f1t = neg ? nega_feats : orig_feats;
  const float* f2t = neg ? nega_feats_pos : orig_feats_pos;
  const float* ct  = neg ? nega_code : orig_code;
  const float* crd = neg ? coords2 : coords1;

  const int b = p / (CCL_S * CCL_S);

  // grid_sample, align_corners=True, border padding.
  // gx = 2u-1 ; x = clip((gx+1)*0.5*(W-1), 0, W-1) = clip(u*(W-1), 0, W-1)
  const float u = crd[2 * p + 0];
  const float v = crd[2 * p + 1];
  const float x = fminf(fmaxf(u * (float)(CCL_W - 1), 0.0f), (float)(CCL_W - 1));
  const float y = fminf(fmaxf(v * (float)(CCL_H - 1), 0.0f), (float)(CCL_H - 1));
  const float x0f = floorf(x);
  const float y0f = floorf(y);
  const int x0 = (int)x0f;
  const int y0 = (int)y0f;
  const int y1 = min(y0 + 1, CCL_H - 1);
  const float wx = x - x0f;
  const float wy = y - y0f;
  const float w00 = (1.0f - wx) * (1.0f - wy);
  const float w01 = wx * (1.0f - wy);
  const float w10 = (1.0f - wx) * wy;
  const float w11 = wx * wy;

  // Pair-gather base: taps (x0, x0+1) as one 64-bit load per row. At the right
  // border (x0 == W-1, where wx == 0 so the x1 tap weight is already 0) shift
  // the pair left one and fold the selection into adjusted weights.
  const int xp = min(x0, CCL_W - 2);
  const bool edge = (x0 == CCL_W - 1);
  const float aw00 = edge ? 0.0f : w00;
  const float aw01 = edge ? (w00 + w01) : w01;
  const float aw10 = edge ? 0.0f : w10;
  const float aw11 = edge ? (w10 + w11) : w11;

  const int o0 = y0 * CCL_W + xp;
  const int o1 = y1 * CCL_W + xp;

  // code sample ([B,1,H,W]) -> clamp(0, 0.8)  (uniform across the wave)
  const float* cb = ct + (size_t)b * CCL_HW;
  const f2u cr0 = *(const f2u*)(cb + o0);
  const f2u cr1 = *(const f2u*)(cb + o1);
  const float code = cr0.x * aw00 + cr0.y * aw01 + cr1.x * aw10 + cr1.y * aw11;
  const float cd = fminf(fmaxf(code, 0.0f), 0.8f);

  // 512 channels: lane handles c = lane + 32*k, k = 0..15; keep interpolated
  // values in registers so the L1 pass needs no re-gather.
  const float* p1 = f1t + (size_t)b * CCL_C * CCL_HW;
  const float* p2 = f2t + (size_t)b * CCL_C * CCL_HW;

  float f1v[16], f2v[16];
  float s1 = 0.0f, s2 = 0.0f;
#pragma unroll
  for (int k = 0; k < 16; ++k) {
    const int c = lane + 32 * k;
    const float* q1 = p1 + (size_t)c * CCL_HW;
    const float* q2 = p2 + (size_t)c * CCL_HW;
    const f2u r10 = *(const f2u*)(q1 + o0);
    const f2u r11 = *(const f2u*)(q1 + o1);
    const f2u r20 = *(const f2u*)(q2 + o0);
    const f2u r21 = *(const f2u*)(q2 + o1);
    const float a1 = fmaf(r10.x, aw00, fmaf(r10.y, aw01, fmaf(r11.x, aw10, r11.y * aw11)));
    const float a2 = fmaf(r20.x, aw00, fmaf(r20.y, aw01, fmaf(r21.x, aw10, r21.y * aw11)));
    f1v[k] = a1;
    f2v[k] = a2;
    s1 = fmaf(a1, a1, s1);
    s2 = fmaf(a2, a2, s2);
  }

  // wave-wide L2 norms via WMMA reductions
  const float S1 = wave_reduce_wmma(s1);
  const float S2 = wave_reduce_wmma(s2);
  const float inv1 = 1.0f / fmaxf(sqrtf(S1), 1e-12f);
  const float inv2 = 1.0f / fmaxf(sqrtf(S2), 1e-12f);

  float dsum = 0.0f;
#pragma unroll
  for (int k = 0; k < 16; ++k) {
    dsum += fabsf(f1v[k] * inv1 - f2v[k] * inv2);
  }
  const float f12 = wave_reduce_wmma(dsum);

  const float fd = tanhf(10.0f * logf(f12 / (1.0f - f12)));
  const float val = cd * fd;
  const float wgt = (neg ? NEG_W : POS_W) / (float)CCL_NPTS;

  if (lane == 0) {
    ws[wid] = val * wgt;
  }
}

__global__ __launch_bounds__(256) void ccl_reduce_kernel(const float* __restrict__ ws,
                                                         float* __restrict__ out, int n) {
  __shared__ float sdata[256];
  float s = 0.0f;
  for (int i = (int)threadIdx.x; i < n; i += 256) s += ws[i];
  sdata[threadIdx.x] = s;
  __syncthreads();
  for (int off = 128; off > 0; off >>= 1) {
    if ((int)threadIdx.x < off) sdata[threadIdx.x] += sdata[threadIdx.x + off];
    __syncthreads();
  }
  if (threadIdx.x == 0) out[0] = sdata[0];
}

extern "C" void kernel_launch(void* const* d_in, const int* in_sizes, int n_in,
                              void* d_out, int out_size, void* d_ws, size_t ws_size,
                              hipStream_t stream) {
  (void)in_sizes; (void)n_in; (void)out_size; (void)ws_size;
  const float* orig_feats     = (const float*)d_in[0];
  const float* orig_feats_pos = (const float*)d_in[1];
  const float* orig_code      = (const float*)d_in[2];
  const float* nega_feats     = (const float*)d_in[3];
  const float* nega_feats_pos = (const float*)d_in[4];
  const float* nega_code      = (const float*)d_in[5];
  const float* coords1        = (const float*)d_in[6];
  const float* coords2        = (const float*)d_in[7];
  float* ws = (float*)d_ws;

  // 7744 waves, 8 waves (256 threads) per block -> 968 blocks, every wave fully
  // populated (EXEC all-ones, required for WMMA).
  ccl_point_kernel<<<CCL_TOTAL / 8, 256, 0, stream>>>(
      orig_feats, orig_feats_pos, orig_code,
      nega_feats, nega_feats_pos, nega_code,
      coords1, coords2, ws);
  ccl_reduce_kernel<<<1, 256, 0, stream>>>(ws, (float*)d_out, CCL_TOTAL);
}